// Attention_13786845020558
// MI455X (gfx1250) — compile-verified
//
#include <hip/hip_runtime.h>
#include <math.h>

typedef __attribute__((ext_vector_type(2))) float v2f;
typedef __attribute__((ext_vector_type(8))) float v8f;

constexpr int kH  = 1024;   // hidden
constexpr int kB  = 32;     // batch
constexpr int kT  = 1024;   // timesteps
constexpr int kK  = 2048;   // 2H (GEMM K)
constexpr int BM  = 128;
constexpr int BN  = 64;
constexpr int BK  = 32;
constexpr int LDP = 34;     // padded LDS row stride (floats), even for b64 loads

__device__ __forceinline__ float selu_f(float x) {
    const float alpha = 1.6732632423543772f;
    const float scale = 1.0507009873554805f;
    return scale * (x > 0.0f ? x : alpha * (expf(x) - 1.0f));
}

// Fused: selu(concat(E, hid) @ W1 + b1) @ W2 -> partial scores (atomic accumulate).
// 128x64 block tile, 8 waves, each wave computes a 32x32 patch (2x2 WMMA subtiles).
__global__ __launch_bounds__(256) void attn_gemm_score(
    const float* __restrict__ E,    // (B,T,H)
    const float* __restrict__ hid,  // (1,B,H)
    const float* __restrict__ W1,   // (2H,H) row-major
    const float* __restrict__ bias1,// (H)
    const float* __restrict__ W2,   // (H,1)
    float* __restrict__ scores)     // (B*T) pre-zeroed
{
    __shared__ float Als[BM * LDP];
    __shared__ float Bls[BN * LDP]; // W1 tile stored transposed: [n][k]

    const int tid  = threadIdx.x;
    const int wave = tid >> 5;
    const int lane = tid & 31;
    const int l15  = lane & 15;
    const int lhi  = lane >> 4;     // half-wave select
    const int wm   = wave & 3;      // M 32-row band 0..3
    const int wn2  = wave >> 2;     // N 32-col band 0..1

    const int m0 = blockIdx.x * BM;
    const int n0 = blockIdx.y * BN;

    // Per-thread staging coords (A: 4 float4, B: 2 float4)
    int ar[4], ac[4];
    #pragma unroll
    for (int i = 0; i < 4; ++i) {
        int idx = tid + i * 256;          // 0..1023
        ar[i] = idx >> 3;                 // 0..127
        ac[i] = (idx & 7) << 2;           // 0,4,...,28
    }
    int bkr[2], bc[2];
    #pragma unroll
    for (int i = 0; i < 2; ++i) {
        int idx = tid + i * 256;          // 0..511
        bkr[i] = idx >> 4;                // 0..31
        bc[i]  = (idx & 15) << 2;         // 0,4,...,60
    }

    float4 pa[4];
    float4 pb[2];

    // prime the pipeline: fetch tile k0=0 into registers
    {
        const int k0 = 0;
        #pragma unroll
        for (int i = 0; i < 4; ++i)
            pa[i] = *(const float4*)(E + (size_t)(m0 + ar[i]) * kH + (k0 + ac[i]));
        #pragma unroll
        for (int i = 0; i < 2; ++i)
            pb[i] = *(const float4*)(W1 + (size_t)(k0 + bkr[i]) * kH + (n0 + bc[i]));
    }

    v8f c00 = {}, c01 = {}, c10 = {}, c11 = {};

    for (int k0 = 0; k0 < kK; k0 += BK) {
        // ---- commit prefetched tile to LDS ----
        #pragma unroll
        for (int i = 0; i < 4; ++i) {
            float* dst = &Als[ar[i] * LDP + ac[i]];
            dst[0] = pa[i].x; dst[1] = pa[i].y; dst[2] = pa[i].z; dst[3] = pa[i].w;
        }
        #pragma unroll
        for (int i = 0; i < 2; ++i) {
            Bls[(bc[i] + 0) * LDP + bkr[i]] = pb[i].x;
            Bls[(bc[i] + 1) * LDP + bkr[i]] = pb[i].y;
            Bls[(bc[i] + 2) * LDP + bkr[i]] = pb[i].z;
            Bls[(bc[i] + 3) * LDP + bkr[i]] = pb[i].w;
        }
        __syncthreads();

        // ---- prefetch next tile (overlaps the WMMA loop below) ----
        const int kn = k0 + BK;
        if (kn < kK) {
            if (kn < kH) {
                #pragma unroll
                for (int i = 0; i < 4; ++i)
                    pa[i] = *(const float4*)(E + (size_t)(m0 + ar[i]) * kH + (kn + ac[i]));
            } else {
                #pragma unroll
                for (int i = 0; i < 4; ++i)
                    pa[i] = *(const float4*)(hid + (size_t)((m0 + ar[i]) / kT) * kH +
                                             (kn - kH + ac[i]));
            }
            #pragma unroll
            for (int i = 0; i < 2; ++i)
                pb[i] = *(const float4*)(W1 + (size_t)(kn + bkr[i]) * kH + (n0 + bc[i]));
        }

        // ---- WMMA inner loop: f32 16x16x4, 2x2 register blocking ----
        // A frag: lane l -> row m=l&15, K pair at kk + 2*(l>=16)
        // B frag: lane l -> col n=l&15, K pair at kk + 2*(l>=16)  (Bls is [n][k])
        #pragma unroll
        for (int kk = 0; kk < BK; kk += 4) {
            const int ko = kk + (lhi << 1);
            v2f a0 = *(const v2f*)&Als[(wm * 32 + l15) * LDP + ko];
            v2f a1 = *(const v2f*)&Als[(wm * 32 + 16 + l15) * LDP + ko];
            v2f b0 = *(const v2f*)&Bls[(wn2 * 32 + l15) * LDP + ko];
            v2f b1 = *(const v2f*)&Bls[(wn2 * 32 + 16 + l15) * LDP + ko];
            c00 = __builtin_amdgcn_wmma_f32_16x16x4_f32(false, a0, false, b0,
                                                        (short)0, c00, false, false);
            c01 = __builtin_amdgcn_wmma_f32_16x16x4_f32(false, a0, false, b1,
                                                        (short)0, c01, false, false);
            c10 = __builtin_amdgcn_wmma_f32_16x16x4_f32(false, a1, false, b0,
                                                        (short)0, c10, false, false);
            c11 = __builtin_amdgcn_wmma_f32_16x16x4_f32(false, a1, false, b1,
                                                        (short)0, c11, false, false);
        }
        __syncthreads();
    }

    // ---- epilogue: bias + SELU + dot(W2) + 16-lane reduce + atomic score ----
    // C layout: element (m,n): n = l&15, m = vgpr + 8*(l>=16)
    const int nc0 = n0 + wn2 * 32 + l15;
    const int nc1 = nc0 + 16;
    const float bia0 = bias1[nc0];
    const float bia1 = bias1[nc1];
    const float w20  = W2[nc0];
    const float w21  = W2[nc1];
    #pragma unroll
    for (int mi = 0; mi < 2; ++mi) {
        const v8f& d0 = mi ? c10 : c00;
        const v8f& d1 = mi ? c11 : c01;
        const int rowBase = m0 + wm * 32 + mi * 16 + (lhi << 3);
        #pragma unroll
        for (int r = 0; r < 8; ++r) {
            float x0 = selu_f(d0[r] + bia0);
            float x1 = selu_f(d1[r] + bia1);
            float p  = x0 * w20 + x1 * w21;
            p += __shfl_xor(p, 1, 32);
            p += __shfl_xor(p, 2, 32);
            p += __shfl_xor(p, 4, 32);
            p += __shfl_xor(p, 8, 32);   // reduced within each 16-lane half
            if (l15 == 0) atomicAdd(&scores[rowBase + r], p);
        }
    }
}

// Per-batch softmax over T (b2 dropped: softmax is shift-invariant) + context.
__global__ __launch_bounds__(256) void attn_softmax_ctx(
    const float* __restrict__ E,       // (B,T,H)
    const float* __restrict__ scores,  // (B*T)
    float* __restrict__ out)           // (1,B,H)
{
    __shared__ float sw[kT];
    __shared__ float red[256];
    const int b   = blockIdx.x;
    const int tid = threadIdx.x;

    float lmax = -INFINITY;
    #pragma unroll
    for (int i = 0; i < 4; ++i) {
        int t = tid + i * 256;
        float v = scores[b * kT + t];
        sw[t] = v;
        lmax = fmaxf(lmax, v);
    }
    red[tid] = lmax;
    __syncthreads();
    for (int s = 128; s > 0; s >>= 1) {
        if (tid < s) red[tid] = fmaxf(red[tid], red[tid + s]);
        __syncthreads();
    }
    const float mx = red[0];
    __syncthreads();

    float lsum = 0.0f;
    #pragma unroll
    for (int i = 0; i < 4; ++i) {
        int t = tid + i * 256;
        float e = expf(sw[t] - mx);
        sw[t] = e;
        lsum += e;
    }
    red[tid] = lsum;
    __syncthreads();
    for (int s = 128; s > 0; s >>= 1) {
        if (tid < s) red[tid] += red[tid + s];
        __syncthreads();
    }
    const float inv = 1.0f / red[0];
    __syncthreads();

    // context[b, 4*tid .. 4*tid+3] = sum_t w[t] * E[b,t,:]
    const float4* E4 = (const float4*)(E + (size_t)b * kT * kH);
    float4 acc = {0.f, 0.f, 0.f, 0.f};
    for (int t = 0; t < kT; ++t) {
        float wt = sw[t] * inv;
        float4 ev = E4[(size_t)t * (kH / 4) + tid];
        acc.x += wt * ev.x;
        acc.y += wt * ev.y;
        acc.z += wt * ev.z;
        acc.w += wt * ev.w;
    }
    ((float4*)out)[b * (kH / 4) + tid] = acc;
}

__global__ void zero_scores(float* __restrict__ s, int n) {
    int i = blockIdx.x * blockDim.x + threadIdx.x;
    if (i < n) s[i] = 0.0f;
}

extern "C" void kernel_launch(void* const* d_in, const int* in_sizes, int n_in,
                              void* d_out, int out_size, void* d_ws, size_t ws_size,
                              hipStream_t stream) {
    const float* E    = (const float*)d_in[0];  // (B,T,H)
    const float* hid  = (const float*)d_in[1];  // (1,B,H)
    const float* W1   = (const float*)d_in[2];  // (2H,H)
    const float* b1   = (const float*)d_in[3];  // (H)
    const float* W2   = (const float*)d_in[4];  // (H,1)
    // d_in[5] = b2: unused (softmax over T is shift-invariant)
    float* out = (float*)d_out;

    // scores scratch: B*T floats. Use d_ws if big enough; otherwise reuse d_out
    // (same flat size B*T == B*H; kernel2 reads its block's scores into LDS
    //  before overwriting that region).
    float* scores = (ws_size >= (size_t)(kB * kT) * sizeof(float)) ? (float*)d_ws
                                                                   : out;

    zero_scores<<<dim3((kB * kT) / 256), 256, 0, stream>>>(scores, kB * kT);
    attn_gemm_score<<<dim3((kB * kT) / BM, kH / BN), 256, 0, stream>>>(
        E, hid, W1, b1, W2, scores);
    attn_softmax_ctx<<<dim3(kB), 256, 0, stream>>>(E, scores, out);
}